// NaNConv2d_75385265980146
// MI455X (gfx1250) — compile-verified
//
#include <hip/hip_runtime.h>

// ---------------------------------------------------------------------------
// NaN-aware 3x3 conv for MI455X (gfx1250), implicit-im2col GEMM on the fp32
// WMMA path (V_WMMA_F32_16X16X4_F32). wave32, 8 waves / 256-thread block.
// Weights stream via CDNA5 async global->LDS DMA (ASYNCcnt), double-buffered.
// ---------------------------------------------------------------------------

typedef float v2f __attribute__((ext_vector_type(2)));
typedef float v8f __attribute__((ext_vector_type(8)));

constexpr int Bn    = 16;
constexpr int Cin   = 64;
constexpr int Hh    = 128;
constexpr int Wh    = 128;
constexpr int Cout  = 128;
constexpr int Ktot  = Cin * 9;          // 576
constexpr int CIPC  = 4;                // input channels per K chunk
constexpr int KC    = CIPC * 9;         // 36 K values per chunk
constexpr int NCH   = Cin / CIPC;       // 16 chunks
constexpr int LDSK  = 38;               // padded K stride (even -> 8B aligned)
constexpr float THRESH_CNT = 0.5f * (Bn * Cin * 9);   // 4608

// CDNA5 async global->LDS copy (VGLOBAL GLOBAL_LOAD_ASYNC_TO_LDS_B64).
// VDST = per-lane LDS byte address (AS3 ptr), VADDR = 64-bit global address.
__device__ __forceinline__ void async_copy_b64(float* lds_dst, const float* gsrc) {
  asm volatile("global_load_async_to_lds_b64 %0, %1, off"
               :
               : "v"((__attribute__((address_space(3))) void*)lds_dst),
                 "v"((const __attribute__((address_space(1))) void*)gsrc)
               : "memory");
}
// Wait until at most N async ops remain outstanding (in-order completion).
#define WAIT_ASYNC(n) asm volatile("s_wait_asynccnt %0" ::"n"(n) : "memory")

// ---- pass 1: NaN count per (h,w), reduced over batch & channels -----------
__global__ __launch_bounds__(256) void nan_count_kernel(
    const float* __restrict__ img, float* __restrict__ msum) {
  int p = blockIdx.x * 256 + threadIdx.x;          // pixel 0..16383
  float s = 0.f;
  #pragma unroll 8
  for (int plane = 0; plane < Bn * Cin; ++plane) { // 1024 coalesced planes
    float v = img[plane * (Hh * Wh) + p];
    s += (v != v) ? 1.f : 0.f;
  }
  msum[p] = s;
}

// ---- pass 2: 3x3 box sum (zero pad) + threshold -> flag map ---------------
__global__ __launch_bounds__(256) void nan_flag_kernel(
    const float* __restrict__ msum, float* __restrict__ flag) {
  int p = blockIdx.x * 256 + threadIdx.x;
  int h = p >> 7, w = p & 127;
  float s = 0.f;
  #pragma unroll
  for (int dh = -1; dh <= 1; ++dh)
    #pragma unroll
    for (int dw = -1; dw <= 1; ++dw) {
      int hh = h + dh, ww = w + dw;
      if ((unsigned)hh < (unsigned)Hh && (unsigned)ww < (unsigned)Wh)
        s += msum[hh * Wh + ww];
    }
  flag[p] = (s >= THRESH_CNT) ? 1.f : 0.f;  // 0.5 * (16*64*9)
}

// ---- pass 3: implicit-GEMM conv via fp32 WMMA -----------------------------
// One workgroup per (b, oh): OUT[Cout=128, OW=128] = Wmat[128,K] x Im2col[K,128]
__global__ __launch_bounds__(256) void nanconv_wmma_kernel(
    const float* __restrict__ img, const float* __restrict__ wgt,
    const float* __restrict__ bias, const float* __restrict__ flag,
    float* __restrict__ out) {
  __shared__ float Wl[2][Cout * LDSK];  // A operand: [co][k], double-buffered
  __shared__ float Xl[Wh * LDSK];       // B operand: [ow][k]

  const int b    = blockIdx.x >> 7;
  const int oh   = blockIdx.x & 127;
  const int tid  = threadIdx.x;
  const int wave = tid >> 5;
  const int lane = tid & 31;
  const int half = lane >> 4;         // selects K pair (A/B) and M offset (C/D)
  const int l16  = lane & 15;
  const int m0   = (wave & 3) * 32;   // co base: 2 M-tiles of 16
  const int n0   = (wave >> 2) * 64;  // ow base: 4 N-tiles of 16

  v8f acc[2][4];
  const v8f vzero = {0.f, 0.f, 0.f, 0.f, 0.f, 0.f, 0.f, 0.f};
  #pragma unroll
  for (int mt = 0; mt < 2; ++mt)
    #pragma unroll
    for (int nt = 0; nt < 4; ++nt)
      acc[mt][nt] = vzero;

  // Prologue: async-DMA weight chunk 0 into Wl[0] (9 b64 ops per thread/wave).
  #pragma unroll
  for (int i = tid; i < (Cout * KC) / 2; i += 256) {
    int kp = i % (KC / 2);            // b64 pair index within a co row
    int co = i / (KC / 2);
    async_copy_b64(&Wl[0][co * LDSK + kp * 2], &wgt[co * Ktot + kp * 2]);
  }

  for (int c = 0; c < NCH; ++c) {
    const int ci0 = c * CIPC;
    const int buf = c & 1;

    // Stage im2col chunk with fused nan_to_num and zero padding.
    // k = (ci-ci0)*9 + kh*3 + kw ; Xl[ow][k]. (Xl safe: barrier at loop end.)
    for (int idx = tid; idx < Wh * KC; idx += 256) {   // 4608 = 18*256, uniform
      int ow = idx & 127;
      int k  = idx >> 7;
      int cq = k / 9;
      int r9 = k - cq * 9;
      int kh = r9 / 3;
      int kw = r9 - kh * 3;
      int ih = oh - 1 + kh;
      int iw = ow - 1 + kw;
      float v = 0.f;
      if ((unsigned)ih < (unsigned)Hh && (unsigned)iw < (unsigned)Wh) {
        v = img[((b * Cin + (ci0 + cq)) * Hh + ih) * Wh + iw];
        v = (v == v) ? v : 0.f;                       // NaN -> 0
      }
      Xl[ow * LDSK + k] = v;
    }

    if (c + 1 < NCH) {
      // Pipeline: async-DMA next weight chunk into the other buffer while
      // this chunk computes. Also warm next image rows (global_prefetch_b8).
      const int ci1 = ci0 + CIPC;
      #pragma unroll
      for (int i = tid; i < (Cout * KC) / 2; i += 256) {
        int kp = i % (KC / 2);
        int co = i / (KC / 2);
        async_copy_b64(&Wl[buf ^ 1][co * LDSK + kp * 2],
                       &wgt[co * Ktot + ci1 * 9 + kp * 2]);
      }
      if (tid < 48) {                 // 12 rows x 4 cachelines of next chunk
        int rq = tid >> 2;
        int cq = rq / 3, khh = rq - (rq / 3) * 3;
        int ih = oh - 1 + khh;
        if ((unsigned)ih < (unsigned)Hh)
          __builtin_prefetch(
              &img[((b * Cin + (ci1 + cq)) * Hh + ih) * Wh + (tid & 3) * 32],
              0, 1);
      }
      WAIT_ASYNC(9);   // current chunk's 9 oldest async ops retired (in-order)
    } else {
      WAIT_ASYNC(0);   // drain: last chunk's weights landed
    }
    __syncthreads();   // all waves' Xl stores + async LDS writes visible

    // Inner GEMM: 9 K-steps of 4, 8 WMMAs each (2 M-tiles x 4 N-tiles).
    #pragma unroll
    for (int k0 = 0; k0 < KC; k0 += 4) {
      v2f a[2], bb[4];
      // A frag (16x4 f32): lanes<->M(co), half-wave selects K pair {0,1}/{2,3}
      #pragma unroll
      for (int mt = 0; mt < 2; ++mt)
        a[mt] = *(const v2f*)&Wl[buf][(m0 + mt * 16 + l16) * LDSK + k0 + 2 * half];
      // B frag (4x16 f32): lanes<->N(ow), half-wave selects K pair
      #pragma unroll
      for (int nt = 0; nt < 4; ++nt)
        bb[nt] = *(const v2f*)&Xl[(n0 + nt * 16 + l16) * LDSK + k0 + 2 * half];

      #pragma unroll
      for (int mt = 0; mt < 2; ++mt)
        #pragma unroll
        for (int nt = 0; nt < 4; ++nt)
          acc[mt][nt] = __builtin_amdgcn_wmma_f32_16x16x4_f32(
              false, a[mt], false, bb[nt], (short)0, acc[mt][nt], false, false);
    }
    __syncthreads();   // compute done: Xl (and retired W buffer) reusable
  }

  // Epilogue: + bias, NaN-threshold override, coalesced stores along ow.
  const float NaNv = __builtin_nanf("");
  #pragma unroll
  for (int nt = 0; nt < 4; ++nt) {
    int ow = n0 + nt * 16 + l16;                 // lane <-> N
    float fl = flag[oh * Wh + ow];
    #pragma unroll
    for (int mt = 0; mt < 2; ++mt) {
      #pragma unroll
      for (int r = 0; r < 8; ++r) {
        int co = m0 + mt * 16 + half * 8 + r;    // C/D: VGPR r -> M = r + 8*half
        float v = acc[mt][nt][r] + bias[co];
        v = (fl != 0.f) ? NaNv : v;
        out[((b * Cout + co) * Hh + oh) * Wh + ow] = v;
      }
    }
  }
}

// ---------------------------------------------------------------------------
extern "C" void kernel_launch(void* const* d_in, const int* in_sizes, int n_in,
                              void* d_out, int out_size, void* d_ws, size_t ws_size,
                              hipStream_t stream) {
  const float* img  = (const float*)d_in[0];   // [16,64,128,128]
  const float* wgt  = (const float*)d_in[1];   // [128,64,3,3]
  const float* bias = (const float*)d_in[2];   // [128]
  float* out = (float*)d_out;                  // [16,128,128,128]

  float* msum = (float*)d_ws;                  // [128*128] NaN counts
  float* flag = msum + Hh * Wh;                // [128*128] threshold flags

  nan_count_kernel<<<(Hh * Wh) / 256, 256, 0, stream>>>(img, msum);
  nan_flag_kernel<<<(Hh * Wh) / 256, 256, 0, stream>>>(msum, flag);
  nanconv_wmma_kernel<<<Bn * Hh, 256, 0, stream>>>(img, wgt, bias, flag, out);
}